// GNNModel_27118423507313
// MI455X (gfx1250) — compile-verified
//
#include <hip/hip_runtime.h>

typedef __attribute__((ext_vector_type(2))) float v2f;
typedef __attribute__((ext_vector_type(4))) float v4f;
typedef __attribute__((ext_vector_type(8))) float v8f;

// ---------------------------------------------------------------------------
// Degree / normalization
// ---------------------------------------------------------------------------
__global__ void k_init_deg(float* __restrict__ deg, int N) {
  int i = blockIdx.x * blockDim.x + threadIdx.x;
  if (i < N) deg[i] = 1.0f;  // self-loop contributes 1 to every node's degree
}

__global__ void k_deg_accum(float* __restrict__ deg, const int* __restrict__ dst, int E) {
  int e = blockIdx.x * blockDim.x + threadIdx.x;
  if (e < E) atomicAdd(&deg[dst[e]], 1.0f);
}

__global__ void k_dinv(float* __restrict__ deg, int N) {
  int i = blockIdx.x * blockDim.x + threadIdx.x;
  if (i < N) {
    float d = deg[i];
    deg[i] = (d > 0.0f) ? rsqrtf(d) : 0.0f;
  }
}

// ---------------------------------------------------------------------------
// Bias pre-initialization of aggregation buffers: g[i*D + c] = bias[c]
// ---------------------------------------------------------------------------
template <int D>
__global__ void k_bias_init(float* __restrict__ g, const float* __restrict__ bias, int total) {
  int t = blockIdx.x * blockDim.x + threadIdx.x;
  if (t < total) g[t] = bias[t & (D - 1)];
}

// ---------------------------------------------------------------------------
// Edge aggregation: one lane per (edge, channel).
// idx in [0,E)    : real edge, w = dinv[src]*dinv[dst]
// idx in [E,E+N)  : self-loop, w = dinv[i]^2
// Gathers are 64B-contiguous per 16-lane group; scatter via
// GLOBAL_ATOMIC_ADD_F32 (no return) into the L2-resident buffer.
// ---------------------------------------------------------------------------
template <int D>
__global__ void k_aggregate(const float* __restrict__ h, const int* __restrict__ src,
                            const int* __restrict__ dst, const float* __restrict__ dinv,
                            float* __restrict__ g, int E, int N) {
  int t = blockIdx.x * blockDim.x + threadIdx.x;
  int total = (E + N) * D;
  if (t >= total) return;
  int c = t & (D - 1);
  int idx = t / D;
  int s, d;
  float w;
  if (idx < E) {
    s = src[idx];
    d = dst[idx];
    w = dinv[s] * dinv[d];
  } else {
    s = idx - E;
    d = s;
    float q = dinv[s];
    w = q * q;
  }
  atomicAdd(&g[d * D + c], h[s * D + c] * w);
}

// ---------------------------------------------------------------------------
// Layer 1 GEMM: h1[N,16] = x[N,3] @ W1[3,16]   (K=3 zero-padded to 4)
// One V_WMMA_F32_16X16X4_F32 per 16-node tile, one wave per tile.
// A 16x4 fp32: lane l holds A[l&15][ (l>>4)*2 + {0,1} ] in {a.x,a.y}
// B 4x16 fp32: lane l holds B[ (l>>4)*2 + {0,1} ][ l&15 ]
// D 16x16 fp32: lane l, vgpr j holds D[ j + 8*(l>>4) ][ l&15 ]
// Fast path (full 16-row tile) is branch-free: K-padding via v_cndmask, not
// exec-masked loads; stores unguarded.
// ---------------------------------------------------------------------------
__global__ void k_gemm1(const float* __restrict__ x, const float* __restrict__ W1,
                        float* __restrict__ h1, int N) {
  int wave = (blockIdx.x * blockDim.x + threadIdx.x) >> 5;
  int lane = threadIdx.x & 31;
  int tiles = (N + 15) >> 4;
  if (wave >= tiles) return;  // wave-uniform; EXEC all-1s below
  int R = wave << 4;
  int n = lane & 15;
  int hh = lane >> 4;   // 0: K={0,1}, 1: K={2,3}
  int row = R + n;
  int k0 = hh * 2;

  // B fragment (always in-bounds, branch-free zero padding for K=3)
  v2f b;
  b.x = W1[k0 * 16 + n];              // K=0 or K=2 row
  float by = W1[16 + n];              // K=1 row (valid load for all lanes)
  b.y = hh ? 0.0f : by;               // K=3 row is zero padding

  v2f a;
  v8f c = {};
  if (R + 16 <= N) {
    // ---- full tile: branch-free ----
    a.x = x[row * 3 + k0];            // K=0 / K=2 (row*3+2 <= 3N-1, in bounds)
    float ay = x[row * 3 + 1];        // K=1 (in bounds for all lanes)
    a.y = hh ? 0.0f : ay;             // K=3 zero padding
    c = __builtin_amdgcn_wmma_f32_16x16x4_f32(false, a, false, b, (short)0, c, false, false);
#pragma unroll
    for (int j = 0; j < 8; j++) {
      int r = R + j + hh * 8;
      h1[r * 16 + n] = c[j];
    }
  } else {
    // ---- tail tile: guarded ----
    bool rv = (row < N);
    a.x = rv ? x[row * 3 + k0] : 0.0f;
    a.y = (rv && !hh) ? x[row * 3 + 1] : 0.0f;
    c = __builtin_amdgcn_wmma_f32_16x16x4_f32(false, a, false, b, (short)0, c, false, false);
#pragma unroll
    for (int j = 0; j < 8; j++) {
      int r = R + j + hh * 8;
      if (r < N) h1[r * 16 + n] = c[j];
    }
  }
}

// ---------------------------------------------------------------------------
// Layer 2 GEMM: h2[N,32] = relu(g1)[N,16] @ W2[16,32]
// 4 K-steps x 2 N-tiles = 8 WMMAs per 16-node tile; ReLU fused into the
// A-fragment load (aligned v2f = global_load_b64).
// ---------------------------------------------------------------------------
__global__ void k_gemm2(const float* __restrict__ g1, const float* __restrict__ W2,
                        float* __restrict__ h2, int N) {
  int wave = (blockIdx.x * blockDim.x + threadIdx.x) >> 5;
  int lane = threadIdx.x & 31;
  int tiles = (N + 15) >> 4;
  if (wave >= tiles) return;
  int R = wave << 4;
  int n = lane & 15;
  int hh = lane >> 4;
  int row = R + n;

  v8f acc0 = {};
  v8f acc1 = {};
  if (R + 16 <= N) {
    // ---- full tile: branch-free ----
#pragma unroll
    for (int kk = 0; kk < 4; kk++) {
      int k0 = kk * 4 + hh * 2;       // even -> 8B-aligned pair
      v2f a = *(const v2f*)(g1 + row * 16 + k0);
      a.x = fmaxf(a.x, 0.0f);
      a.y = fmaxf(a.y, 0.0f);
      v2f b0, b1;
      b0.x = W2[k0 * 32 + n];
      b0.y = W2[(k0 + 1) * 32 + n];
      b1.x = W2[k0 * 32 + 16 + n];
      b1.y = W2[(k0 + 1) * 32 + 16 + n];
      acc0 = __builtin_amdgcn_wmma_f32_16x16x4_f32(false, a, false, b0, (short)0, acc0, false, false);
      acc1 = __builtin_amdgcn_wmma_f32_16x16x4_f32(false, a, false, b1, (short)0, acc1, false, false);
    }
#pragma unroll
    for (int j = 0; j < 8; j++) {
      int r = R + j + hh * 8;
      h2[r * 32 + n] = acc0[j];
      h2[r * 32 + 16 + n] = acc1[j];
    }
  } else {
    // ---- tail tile: guarded ----
    bool rv = (row < N);
#pragma unroll
    for (int kk = 0; kk < 4; kk++) {
      int k0 = kk * 4 + hh * 2;
      v2f a;
      a.x = rv ? fmaxf(g1[row * 16 + k0], 0.0f) : 0.0f;
      a.y = rv ? fmaxf(g1[row * 16 + k0 + 1], 0.0f) : 0.0f;
      v2f b0, b1;
      b0.x = W2[k0 * 32 + n];
      b0.y = W2[(k0 + 1) * 32 + n];
      b1.x = W2[k0 * 32 + 16 + n];
      b1.y = W2[(k0 + 1) * 32 + 16 + n];
      acc0 = __builtin_amdgcn_wmma_f32_16x16x4_f32(false, a, false, b0, (short)0, acc0, false, false);
      acc1 = __builtin_amdgcn_wmma_f32_16x16x4_f32(false, a, false, b1, (short)0, acc1, false, false);
    }
#pragma unroll
    for (int j = 0; j < 8; j++) {
      int r = R + j + hh * 8;
      if (r < N) {
        h2[r * 32 + n] = acc0[j];
        h2[r * 32 + 16 + n] = acc1[j];
      }
    }
  }
}

// ---------------------------------------------------------------------------
// Layer 3 GEMM: h3[N,2] = relu(g2)[N,32] @ W3[32,2]
// Only 2 output columns -> per-node VALU dot, row loaded as 8x b128.
// ---------------------------------------------------------------------------
__global__ void k_gemm3(const float* __restrict__ g2, const float* __restrict__ W3,
                        float* __restrict__ h3, int N) {
  int i = blockIdx.x * blockDim.x + threadIdx.x;
  if (i >= N) return;
  const v4f* row = (const v4f*)(g2 + (size_t)i * 32);
  float o0 = 0.0f, o1 = 0.0f;
#pragma unroll
  for (int q = 0; q < 8; q++) {
    v4f v = row[q];
#pragma unroll
    for (int u = 0; u < 4; u++) {
      int k = q * 4 + u;
      float r = fmaxf(v[u], 0.0f);
      o0 += r * W3[k * 2 + 0];
      o1 += r * W3[k * 2 + 1];
    }
  }
  h3[i * 2 + 0] = o0;
  h3[i * 2 + 1] = o1;
}

// ---------------------------------------------------------------------------
// Launch
// ---------------------------------------------------------------------------
extern "C" void kernel_launch(void* const* d_in, const int* in_sizes, int n_in,
                              void* d_out, int out_size, void* d_ws, size_t ws_size,
                              hipStream_t stream) {
  const float* x  = (const float*)d_in[0];
  const int*   ei = (const int*)d_in[1];
  const float* W1 = (const float*)d_in[2];
  const float* b1 = (const float*)d_in[3];
  const float* W2 = (const float*)d_in[4];
  const float* b2 = (const float*)d_in[5];
  const float* W3 = (const float*)d_in[6];
  const float* b3 = (const float*)d_in[7];
  float* out = (float*)d_out;

  const int N = in_sizes[0] / 3;
  const int E = in_sizes[1] / 2;
  const int* src = ei;
  const int* dst = ei + E;

  // Workspace layout (floats): dinv[N], h1[N*16], g1[N*16], h2[N*32], g2[N*32], h3[N*2]
  float* ws   = (float*)d_ws;
  float* dinv = ws;
  float* h1   = dinv + (size_t)N;
  float* g1   = h1 + (size_t)N * 16;
  float* h2   = g1 + (size_t)N * 16;
  float* g2   = h2 + (size_t)N * 32;
  float* h3   = g2 + (size_t)N * 32;

  const int B = 256;
  auto cdiv = [](long long a, long long b) { return (int)((a + b - 1) / b); };

  // Normalization: deg -> dinv (in place)
  k_init_deg<<<cdiv(N, B), B, 0, stream>>>(dinv, N);
  k_deg_accum<<<cdiv(E, B), B, 0, stream>>>(dinv, dst, E);
  k_dinv<<<cdiv(N, B), B, 0, stream>>>(dinv, N);

  const int tiles = (N + 15) / 16;
  const int gemmGrid = cdiv((long long)tiles * 32, B);

  // Layer 1
  k_gemm1<<<gemmGrid, B, 0, stream>>>(x, W1, h1, N);
  k_bias_init<16><<<cdiv((long long)N * 16, B), B, 0, stream>>>(g1, b1, N * 16);
  k_aggregate<16><<<cdiv((long long)(E + N) * 16, B), B, 0, stream>>>(h1, src, dst, dinv, g1, E, N);

  // Layer 2 (ReLU of g1 fused into GEMM2 A-load)
  k_gemm2<<<gemmGrid, B, 0, stream>>>(g1, W2, h2, N);
  k_bias_init<32><<<cdiv((long long)N * 32, B), B, 0, stream>>>(g2, b2, N * 32);
  k_aggregate<32><<<cdiv((long long)(E + N) * 32, B), B, 0, stream>>>(h2, src, dst, dinv, g2, E, N);

  // Layer 3 (ReLU of g2 fused into GEMM3; output = agg + b3, no ReLU)
  k_gemm3<<<cdiv(N, B), B, 0, stream>>>(g2, W3, h3, N);
  k_bias_init<2><<<cdiv((long long)N * 2, B), B, 0, stream>>>(out, b3, N * 2);
  k_aggregate<2><<<cdiv((long long)(E + N) * 2, B), B, 0, stream>>>(h3, src, dst, dinv, out, E, N);
}